// dgl_gcn_73529840107893
// MI455X (gfx1250) — compile-verified
//
#include <hip/hip_runtime.h>

typedef __attribute__((ext_vector_type(2))) float v2f;
typedef __attribute__((ext_vector_type(8))) float v8f;

// ---------------------------------------------------------------------------
// Degree accumulation: one thread per edge, 4 float atomics (deg arrays)
// ---------------------------------------------------------------------------
__global__ void degree_kernel(const int* __restrict__ s1, const int* __restrict__ d1,
                              const int* __restrict__ s2, const int* __restrict__ d2,
                              float* __restrict__ so1, float* __restrict__ si1,
                              float* __restrict__ so2, float* __restrict__ si2, int E) {
    int e = blockIdx.x * blockDim.x + threadIdx.x;
    if (e >= E) return;
    unsafeAtomicAdd(&so1[s1[e]], 1.0f);
    unsafeAtomicAdd(&si1[d1[e]], 1.0f);
    unsafeAtomicAdd(&so2[s2[e]], 1.0f);
    unsafeAtomicAdd(&si2[d2[e]], 1.0f);
}

// s[i] = rsqrt(max(deg, 1))  (jnp.clip(deg, 1.0) then rsqrt)
__global__ void finalize_scale_kernel(float* __restrict__ s, int n) {
    int i = blockIdx.x * blockDim.x + threadIdx.x;
    if (i >= n) return;
    s[i] = rsqrtf(fmaxf(s[i], 1.0f));
}

// ---------------------------------------------------------------------------
// WMMA f32 GEMM: O[nrows x NCOL] = X[nrows x K] @ X W[K x NCOL]
// One wave computes a 16-row x NCOL strip via V_WMMA_F32_16X16X4_F32.
//
// W is staged into LDS pre-packed in B-fragment order: for K-group g
// (4 K-rows) and half h, element j of lane (h, col)'s fragment lives at
//   sP[(2g + h)*PS + 2*col + j] = W[4g + 2h + j][col]
// so each lane's B fragment is one aligned ds_load_b64 (v2f).
// PS = 2*NCOL + 32 => the two half-wave bases differ by 32 banks (mod 64),
// so lanes 0-15 and 16-31 hit disjoint bank halves.
// ---------------------------------------------------------------------------
template <int K, int NCOL, int KC>
__global__ __launch_bounds__(128) void gemm_wmma_kernel(const float* __restrict__ X,
                                                        const float* __restrict__ W,
                                                        float* __restrict__ O, int nrows) {
    constexpr int PS = 2 * NCOL + 32;  // packed half-row pitch (floats)
    constexpr int NT = NCOL / 16;      // 16-wide column tiles
    __shared__ float sP[(KC / 2) * PS];

    const int wave = threadIdx.x >> 5;
    const int lane = threadIdx.x & 31;
    const int m    = lane & 15;        // A row / B col / C col within tile
    const int half = lane >> 4;        // 0: K={0,1}, 1: K={2,3}
    const int klo  = half * 2;

    const int  rowBase = (blockIdx.x * 4 + wave) * 16;
    const bool active  = rowBase < nrows;  // uniform per wave (N % 16 == 0)
    const float* xrow  = X + (size_t)(rowBase + m) * K;

    v8f acc[NT];
#pragma unroll
    for (int t = 0; t < NT; ++t)
#pragma unroll
        for (int v = 0; v < 8; ++v) acc[t][v] = 0.0f;

    for (int kb = 0; kb < K; kb += KC) {
        __syncthreads();
        for (int idx = threadIdx.x; idx < KC * NCOL; idx += 128) {
            int r = idx / NCOL, c = idx % NCOL;   // NCOL is a power of two
            int g = r >> 2, h = (r >> 1) & 1, j = r & 1;
            sP[(2 * g + h) * PS + 2 * c + j] = W[(kb + r) * NCOL + c];
        }
        __syncthreads();
        if (active) {
            const float* xk = xrow + kb + klo;
#pragma unroll 4
            for (int g = 0; g < KC / 4; ++g) {
                v2f a = *(const v2f*)(xk + g * 4);              // A: K = 4g+klo, +1
                const float* bp = &sP[(2 * g + half) * PS + 2 * m];
#pragma unroll
                for (int t = 0; t < NT; ++t) {
                    v2f b = *(const v2f*)(bp + t * 32);         // B frag: one b64 load
                    acc[t] = __builtin_amdgcn_wmma_f32_16x16x4_f32(
                        false, a, false, b, (short)0, acc[t], false, false);
                }
            }
        }
    }

    if (active) {
#pragma unroll
        for (int t = 0; t < NT; ++t)
#pragma unroll
            for (int v = 0; v < 8; ++v)   // C/D: row = half*8 + v, col = m
                O[(size_t)(rowBase + half * 8 + v) * NCOL + t * 16 + m] = acc[t][v];
    }
}

// ---------------------------------------------------------------------------
// SpMM scatter: agg[dst[e], :] += s_out[src[e]] * H[src[e], :]
// One thread per (edge, 4-channel chunk): float4 gather + 4 f32 atomics.
// ---------------------------------------------------------------------------
template <int C>
__global__ void scatter_kernel(const float* __restrict__ H, const int* __restrict__ src,
                               const int* __restrict__ dst, const float* __restrict__ s_out,
                               float* __restrict__ agg, unsigned total) {
    constexpr unsigned Q = C / 4;
    unsigned tid = blockIdx.x * blockDim.x + threadIdx.x;
    if (tid >= total) return;
    unsigned e = tid / Q;
    unsigned q = tid % Q;
    int s = src[e];
    int d = dst[e];
    float  sc = s_out[s];
    float4 h  = ((const float4*)H)[(size_t)s * Q + q];
    float* base = &agg[(size_t)d * C + q * 4];
    unsafeAtomicAdd(base + 0, sc * h.x);
    unsafeAtomicAdd(base + 1, sc * h.y);
    unsafeAtomicAdd(base + 2, sc * h.z);
    unsafeAtomicAdd(base + 3, sc * h.w);
}

// ---------------------------------------------------------------------------
// out[i,c] = 0.9*(agg1[i,c]*si1[i] + b[c]) + 0.8*(agg2[i,c]*si2[i] + b[c])
// Vectorized by float4.
// ---------------------------------------------------------------------------
template <int C>
__global__ void combine_kernel(const float4* __restrict__ agg1, const float4* __restrict__ agg2,
                               const float* __restrict__ si1, const float* __restrict__ si2,
                               const float4* __restrict__ bias, float4* __restrict__ out,
                               unsigned total4) {
    constexpr unsigned Q = C / 4;
    unsigned idx = blockIdx.x * blockDim.x + threadIdx.x;
    if (idx >= total4) return;
    unsigned i = idx / Q;
    unsigned c = idx % Q;
    float4 a1 = agg1[idx];
    float4 a2 = agg2[idx];
    float4 b  = bias[c];
    float  s1 = si1[i];
    float  s2 = si2[i];
    float4 o;
    o.x = 0.9f * (a1.x * s1 + b.x) + 0.8f * (a2.x * s2 + b.x);
    o.y = 0.9f * (a1.y * s1 + b.y) + 0.8f * (a2.y * s2 + b.y);
    o.z = 0.9f * (a1.z * s1 + b.z) + 0.8f * (a2.z * s2 + b.z);
    o.w = 0.9f * (a1.w * s1 + b.w) + 0.8f * (a2.w * s2 + b.w);
    out[idx] = o;
}

// ---------------------------------------------------------------------------
extern "C" void kernel_launch(void* const* d_in, const int* in_sizes, int n_in,
                              void* d_out, int out_size, void* d_ws, size_t ws_size,
                              hipStream_t stream) {
    const float* feat = (const float*)d_in[0];   // [N, 256]
    const float* W1   = (const float*)d_in[1];   // [256, 64]
    const float* b1   = (const float*)d_in[2];   // [64]
    const float* W2   = (const float*)d_in[3];   // [64, 32]
    const float* b2   = (const float*)d_in[4];   // [32]
    // d_in[5..12]: gate linears -> mathematically no-ops (softmax over size-1 axis)
    const int* src1 = (const int*)d_in[13];
    const int* dst1 = (const int*)d_in[14];
    const int* src2 = (const int*)d_in[15];
    const int* dst2 = (const int*)d_in[16];

    const int N = in_sizes[0] / 256;
    const int E = in_sizes[13];

    // Workspace layout (floats), with aliasing of dead buffers:
    float* ws   = (float*)d_ws;
    float* so1  = ws;                         // [N]
    float* si1  = so1 + N;                    // [N]
    float* so2  = si1 + N;                    // [N]
    float* si2  = so2 + N;                    // [N]
    float* H    = si2 + N;                    // [N,64]  layer-1 GEMM out; later x1
    float* agg1 = H + (size_t)N * 64;         // [N,64]; later H2
    float* agg2 = agg1 + (size_t)N * 64;      // [N,64]; later agg21|agg22
    float* x1    = H;
    float* H2    = agg1;
    float* agg21 = agg2;
    float* agg22 = agg2 + (size_t)N * 32;

    // --- degrees -> rsqrt scales ---
    hipMemsetAsync(so1, 0, (size_t)4 * N * sizeof(float), stream);
    degree_kernel<<<(E + 255) / 256, 256, 0, stream>>>(src1, dst1, src2, dst2,
                                                       so1, si1, so2, si2, E);
    finalize_scale_kernel<<<(4 * N + 255) / 256, 256, 0, stream>>>(so1, 4 * N);

    // --- layer 1: H = X @ W1 (WMMA), then per-graph scatter ---
    gemm_wmma_kernel<256, 64, 128><<<(N + 63) / 64, 128, 0, stream>>>(feat, W1, H, N);
    hipMemsetAsync(agg1, 0, (size_t)2 * N * 64 * sizeof(float), stream);
    {
        unsigned total = (unsigned)E * 16u;   // E * 64 / 4
        scatter_kernel<64><<<(total + 255) / 256, 256, 0, stream>>>(H, src1, dst1, so1, agg1, total);
        scatter_kernel<64><<<(total + 255) / 256, 256, 0, stream>>>(H, src2, dst2, so2, agg2, total);
    }
    {
        unsigned total4 = (unsigned)N * 16u;  // N * 64 / 4
        combine_kernel<64><<<(total4 + 255) / 256, 256, 0, stream>>>(
            (const float4*)agg1, (const float4*)agg2, si1, si2, (const float4*)b1,
            (float4*)x1, total4);
    }

    // --- layer 2: H2 = x1 @ W2 (WMMA), then per-graph scatter ---
    gemm_wmma_kernel<64, 32, 64><<<(N + 63) / 64, 128, 0, stream>>>(x1, W2, H2, N);
    hipMemsetAsync(agg2, 0, (size_t)2 * N * 32 * sizeof(float), stream);
    {
        unsigned total = (unsigned)E * 8u;    // E * 32 / 4
        scatter_kernel<32><<<(total + 255) / 256, 256, 0, stream>>>(H2, src1, dst1, so1, agg21, total);
        scatter_kernel<32><<<(total + 255) / 256, 256, 0, stream>>>(H2, src2, dst2, so2, agg22, total);
    }
    {
        unsigned total4 = (unsigned)N * 8u;   // N * 32 / 4
        combine_kernel<32><<<(total4 + 255) / 256, 256, 0, stream>>>(
            (const float4*)agg21, (const float4*)agg22, si1, si2, (const float4*)b2,
            (float4*)d_out, total4);
    }
}